// MessagePassingLayer_27127013442152
// MI455X (gfx1250) — compile-verified
//
#include <hip/hip_runtime.h>

typedef __attribute__((ext_vector_type(2))) float v2f;
typedef __attribute__((ext_vector_type(8))) float v8f;

#define IN_DIM  64
#define OUT_DIM 128
#define BN_EPS  1e-5f

// ---------------- degree / dinv ----------------
__global__ void init_deg_kernel(float* __restrict__ deg, int n) {
    int i = blockIdx.x * blockDim.x + threadIdx.x;
    if (i < n) deg[i] = 1.0f;          // self loop contributes 1
}

__global__ void zero_bn_kernel(float* __restrict__ sums) {
    int i = threadIdx.x;
    if (i < 2 * OUT_DIM) sums[i] = 0.0f;
}

__global__ void deg_kernel(const int* __restrict__ col, float* __restrict__ deg, int E) {
    int e = blockIdx.x * blockDim.x + threadIdx.x;
    if (e < E) atomicAdd(&deg[col[e]], 1.0f);
}

__global__ void dinv_kernel(const float* __restrict__ deg, float* __restrict__ dinv, int n) {
    int i = blockIdx.x * blockDim.x + threadIdx.x;
    if (i < n) dinv[i] = rsqrtf(deg[i]);
}

// ---------------- fused dual GEMM via f32 WMMA ----------------
// Each wave computes one 16x16 tile of BOTH xt = x@W and skip = x@skip_W,
// sharing the A fragments. Epilogue writes xt and seeds the output
// accumulator with skip + bias + dinv[row]^2 * xt (self-loop message).
__global__ __launch_bounds__(256)
void gemm_fused_kernel(const float* __restrict__ x, const float* __restrict__ W,
                       const float* __restrict__ skipW, const float* __restrict__ bias,
                       const float* __restrict__ dinv,
                       float* __restrict__ xt, float* __restrict__ outacc, int n) {
    const int lane  = threadIdx.x & 31;
    const int wave  = threadIdx.x >> 5;       // 8 waves -> 8 column tiles
    const int row0  = blockIdx.x * 16;
    const int col0  = wave * 16;
    const int laneN = lane & 15;
    const int hi    = lane >> 4;              // 0: K={0,1}, 1: K={2,3}

    v8f accW = {};
    v8f accS = {};

    // A fragment source row (clamped for the tail tile; WMMA needs full EXEC,
    // clamping keeps loads safe without divergence)
    int rA = row0 + laneN;
    if (rA >= n) rA = n - 1;
    const float* __restrict__ xrow = x + (size_t)rA * IN_DIM + (hi << 1);

    const int nB = col0 + laneN;
#pragma unroll
    for (int kb = 0; kb < IN_DIM / 4; ++kb) {
        const int k0 = kb * 4;
        v2f a;
        a.x = xrow[k0];
        a.y = xrow[k0 + 1];

        const int kB = k0 + (hi << 1);
        v2f bW, bS;
        bW.x = W[(size_t)kB * OUT_DIM + nB];
        bW.y = W[(size_t)(kB + 1) * OUT_DIM + nB];
        bS.x = skipW[(size_t)kB * OUT_DIM + nB];
        bS.y = skipW[(size_t)(kB + 1) * OUT_DIM + nB];

        accW = __builtin_amdgcn_wmma_f32_16x16x4_f32(false, a, false, bW,
                                                     (short)0, accW, false, false);
        accS = __builtin_amdgcn_wmma_f32_16x16x4_f32(false, a, false, bS,
                                                     (short)0, accS, false, false);
    }

    const float bval = bias[nB];
#pragma unroll
    for (int r = 0; r < 8; ++r) {
        const int m   = r + (hi << 3);        // C/D layout: lanes16-31 hold M+8
        const int row = row0 + m;
        if (row < n) {
            const float xv = accW[r];
            const float di = dinv[row];
            xt[(size_t)row * OUT_DIM + nB]     = xv;
            outacc[(size_t)row * OUT_DIM + nB] = accS[r] + bval + di * di * xv;
        }
    }
}

// ---------------- edge scatter (L2 atomic bound) ----------------
__global__ __launch_bounds__(256)
void scatter_kernel(const int* __restrict__ ei, const float* __restrict__ xt,
                    const float* __restrict__ dinv, float* __restrict__ outacc, int E) {
    const long long tid = (long long)blockIdx.x * blockDim.x + threadIdx.x;
    const int e = (int)(tid >> 5);
    if (e >= E) return;
    const int c4 = (int)(tid & 31) * 4;       // 32 lanes x 4 cols = 128
    const int r = ei[e];
    const int c = ei[E + e];
    const float w = dinv[r] * dinv[c];
    const float4 v = *(const float4*)(xt + (size_t)r * OUT_DIM + c4);
    float* dst = outacc + (size_t)c * OUT_DIM + c4;
    atomicAdd(dst + 0, v.x * w);
    atomicAdd(dst + 1, v.y * w);
    atomicAdd(dst + 2, v.z * w);
    atomicAdd(dst + 3, v.w * w);
}

// ---------------- batch-norm statistics ----------------
__global__ __launch_bounds__(128)
void bn_reduce_kernel(const float* __restrict__ acc, float* __restrict__ sums, int n) {
    const int c  = threadIdx.x;               // one column per thread (coalesced rows)
    const int r0 = blockIdx.x * 512;
    int r1 = r0 + 512;
    if (r1 > n) r1 = n;
    float s = 0.0f, s2 = 0.0f;
    for (int r = r0; r < r1; ++r) {
        const float v = acc[(size_t)r * OUT_DIM + c];
        s += v;
        s2 += v * v;
    }
    atomicAdd(&sums[c], s);
    atomicAdd(&sums[OUT_DIM + c], s2);
}

__global__ __launch_bounds__(256)
void bn_finalize_kernel(float* __restrict__ out, const float* __restrict__ sums,
                        const float* __restrict__ gamma, const float* __restrict__ beta,
                        int n) {
    const long long i = (long long)blockIdx.x * blockDim.x + threadIdx.x;
    if (i >= (long long)n * OUT_DIM) return;
    const int c = (int)(i & (OUT_DIM - 1));
    const float inv_n = 1.0f / (float)n;
    const float mean  = sums[c] * inv_n;
    const float var   = sums[OUT_DIM + c] * inv_n - mean * mean;
    float v = (out[i] - mean) * rsqrtf(var + BN_EPS) * gamma[c] + beta[c];
    out[i] = v > 0.0f ? v : 0.0f;
}

extern "C" void kernel_launch(void* const* d_in, const int* in_sizes, int n_in,
                              void* d_out, int out_size, void* d_ws, size_t ws_size,
                              hipStream_t stream) {
    const float* x     = (const float*)d_in[0];
    const int*   ei    = (const int*)d_in[1];
    const float* W     = (const float*)d_in[2];
    const float* bias  = (const float*)d_in[3];
    const float* skipW = (const float*)d_in[4];
    const float* gamma = (const float*)d_in[5];
    const float* beta  = (const float*)d_in[6];

    const int n = in_sizes[0] / IN_DIM;
    const int E = in_sizes[1] / 2;
    float* out = (float*)d_out;

    // workspace layout (floats): xt[n*128] | deg[n] | dinv[n] | bn sums[256]
    float* ws   = (float*)d_ws;
    float* xt   = ws;
    float* deg  = xt + (size_t)n * OUT_DIM;
    float* dinv = deg + n;
    float* sums = dinv + n;

    init_deg_kernel<<<(n + 255) / 256, 256, 0, stream>>>(deg, n);
    zero_bn_kernel<<<1, 256, 0, stream>>>(sums);
    deg_kernel<<<(E + 255) / 256, 256, 0, stream>>>(ei + E, deg, E);
    dinv_kernel<<<(n + 255) / 256, 256, 0, stream>>>(deg, dinv, n);

    gemm_fused_kernel<<<(n + 15) / 16, 256, 0, stream>>>(x, W, skipW, bias, dinv,
                                                         xt, out, n);

    const long long sc_threads = (long long)E * 32;
    scatter_kernel<<<(int)((sc_threads + 255) / 256), 256, 0, stream>>>(ei, xt, dinv,
                                                                        out, E);

    bn_reduce_kernel<<<(n + 511) / 512, 128, 0, stream>>>(out, sums, n);

    const long long elems = (long long)n * OUT_DIM;
    bn_finalize_kernel<<<(int)((elems + 255) / 256), 256, 0, stream>>>(out, sums,
                                                                       gamma, beta, n);
}